// RGCN_DualAttn_FFNN_30416958390829
// MI455X (gfx1250) — compile-verified
//
#include <hip/hip_runtime.h>
#include <hip/hip_bf16.h>

// ---------------------------------------------------------------------------
// RGCN (2 layers) + dual single-query MHA + FFNN + BCE loss for MI455X/gfx1250
//
// MI455X reasoning: workload is memory/atomic bound (205MB per-relation
// accumulator ~= L2 size; GEMMs only ~30 GFLOP), so we keep f32 end-to-end
// and use V_WMMA_F32_16X16X4_F32 for the 9x (50000x128x128) transforms.
// B-operand k-chunks are staged to LDS with GLOBAL_LOAD_ASYNC_TO_LDS_B128
// (gfx1250 async copy) so WMMAs are fed from DS instead of latency-bound
// scattered global b32 loads.
// ---------------------------------------------------------------------------

#define NNODES 50000
#define DIM    128
#define NREL   8
#define BATCHN 1024
#define NHEADS 8
#define NSUBJ  30
#define NCOSP  25
#define BROW   (3 + NSUBJ + NCOSP)   // 58
#define MAXL   30

typedef __attribute__((ext_vector_type(2))) float v2f;
typedef __attribute__((ext_vector_type(8))) float v8f;
typedef __attribute__((ext_vector_type(4))) int   v4i;

#if __has_builtin(__builtin_amdgcn_global_load_async_to_lds_b128)
#define HAVE_ASYNC_LDS 1
// builtin signature (from hipcc diagnostic): param0 = global (AS1) int4*,
// param1 = LDS (AS3) int4*, then imm offset, imm cpol.
typedef __attribute__((address_space(1))) v4i g_v4i;
typedef __attribute__((address_space(3))) v4i l_v4i;
#else
#define HAVE_ASYNC_LDS 0
#endif

static __device__ __forceinline__ void wait_asynccnt0()
{
#if __has_builtin(__builtin_amdgcn_s_wait_asynccnt)
    __builtin_amdgcn_s_wait_asynccnt(0);
#else
    asm volatile("s_wait_asynccnt 0x0" ::: "memory");
#endif
}

// ---------------------------------------------------------------------------
// Kernel 1: scatter-add messages per (relation, dst).  One wave32 per edge;
// each lane moves float4 (32 lanes x 4 = 128 dims) with global f32 atomics.
// ---------------------------------------------------------------------------
__global__ __launch_bounds__(256)
void rgcn_scatter(const float* __restrict__ x,
                  const int*   __restrict__ ei,   // [2, E]: src row then dst row
                  const int*   __restrict__ et,   // [E]
                  float*       __restrict__ acc,  // [NREL, NNODES, DIM]
                  float*       __restrict__ cnt,  // [NREL, NNODES]
                  int E)
{
    int wave = blockIdx.x * (blockDim.x >> 5) + (threadIdx.x >> 5);
    int lane = threadIdx.x & 31;
    if (wave >= E) return;
    int src = ei[wave];
    int dst = ei[E + wave];
    int rel = et[wave];
    const float4 v = *(const float4*)(x + (size_t)src * DIM + lane * 4);
    float* p = acc + ((size_t)rel * NNODES + dst) * DIM + lane * 4;
    atomicAdd(p + 0, v.x);
    atomicAdd(p + 1, v.y);
    atomicAdd(p + 2, v.z);
    atomicAdd(p + 3, v.w);
    if (lane == 0) atomicAdd(cnt + (size_t)rel * NNODES + dst, 1.0f);
}

// ---------------------------------------------------------------------------
// Kernel 2: h_out[n,e] = sum_r mean[r,n,:] @ Wrel[r][:,e] + x[n,:]@Wroot[e,:]^T
//           + b[e]  (optionally ReLU).
// One wave per 16-node x 128-col tile; 4 waves per block.
// A-panel (16x128 f32, pre-scaled by 1/max(cnt,1)) staged in LDS per wave;
// B k-chunks (32x128 f32 = 16KB, contiguous for Wrel) staged in LDS per block
// via async-to-LDS; V_WMMA_F32_16X16X4_F32 accumulation.
// ---------------------------------------------------------------------------
__global__ __launch_bounds__(128)
void rgcn_gemm(const float* __restrict__ x,      // [NNODES, DIM]   layer input
               const float* __restrict__ acc,    // [NREL, NNODES, DIM]
               const float* __restrict__ cnt,    // [NREL, NNODES]
               const float* __restrict__ Wrel,   // [NREL, DIM, DIM]  (d,e)
               const float* __restrict__ Wroot,  // [DIM, DIM]        (e,d)
               const float* __restrict__ bias,   // [DIM]
               float*       __restrict__ hout,   // [NNODES, DIM]
               int relu)
{
    __shared__ __align__(16) float panel[4][16][DIM + 4];  // per-wave A panel
    __shared__ __align__(16) float Bs[32 * DIM];           // B k-chunk, [kk][n]

    const int wave = threadIdx.x >> 5;
    const int lane = threadIdx.x & 31;
    const int tid  = threadIdx.x;
    const int row0 = (blockIdx.x * 4 + wave) * 16;   // first node of this tile

    v8f c[8] = {};                                   // 8 n-tiles of 16x16 f32

    // WMMA f32 16x16x4 lane mapping (ISA 7.12.2):
    //   A: m = lane%16, VGPR v holds K = v + 2*(lane/16)
    //   B: n = lane%16, VGPR v holds K = v + 2*(lane/16)
    //   C: VGPR i -> row m = i + 8*(lane/16), col n = lane%16
    const int mA = lane & 15;
    const int kHalf = (lane >> 4) << 1;              // 0 or 2

    for (int mat = 0; mat < NREL + 1; ++mat) {
        __syncthreads();   // prior panel reads complete before restage

        // ---- stage A panel: 16 rows x 128 cols, 16 float4's per lane ----
        for (int t = lane; t < 16 * (DIM / 4); t += 32) {
            int r    = t >> 5;            // row 0..15      (DIM/4 == 32)
            int cidx = (t & 31) << 2;     // col 0..124 step 4
            int node = row0 + r;
            float4 v = make_float4(0.f, 0.f, 0.f, 0.f);
            if (node < NNODES) {
                if (mat < NREL) {
                    float cc   = cnt[(size_t)mat * NNODES + node];
                    float invc = 1.0f / fmaxf(cc, 1.0f);
                    const float4 a4 = *(const float4*)(acc + ((size_t)mat * NNODES + node) * DIM + cidx);
                    v.x = a4.x * invc; v.y = a4.y * invc; v.z = a4.z * invc; v.w = a4.w * invc;
                } else {
                    v = *(const float4*)(x + (size_t)node * DIM + cidx);
                }
            }
            *(float4*)&panel[wave][r][cidx] = v;
        }

        for (int kc = 0; kc < 4; ++kc) {
            const int ks0 = kc * 32;
            __syncthreads();   // prior Bs reads done; panel visible (kc==0)

            // ---- stage B k-chunk into LDS ----
            if (mat < NREL) {
                // contiguous 16KB block: Wrel[mat][ks0 .. ks0+31][0..127]
                const float* gsrc = Wrel + ((size_t)mat * DIM + ks0) * DIM;
#if HAVE_ASYNC_LDS
#pragma unroll
                for (int i = 0; i < 8; ++i) {
                    int e = (i * 128 + tid) * 4;     // float index, 16B aligned
                    __builtin_amdgcn_global_load_async_to_lds_b128(
                        (g_v4i*)(gsrc + e), (l_v4i*)(Bs + e), 0, 0);
                }
                wait_asynccnt0();
#else
#pragma unroll
                for (int i = 0; i < 8; ++i) {
                    int e = (i * 128 + tid) * 4;
                    *(float4*)(Bs + e) = *(const float4*)(gsrc + e);
                }
#endif
            } else {
                // root: Bs[kk][n] = Wroot[n][ks0+kk], transpose while staging
                const float* wr = Wroot + (size_t)tid * DIM + ks0;
#pragma unroll
                for (int kk = 0; kk < 32; kk += 4) {
                    float4 v = *(const float4*)(wr + kk);
                    Bs[(kk + 0) * DIM + tid] = v.x;
                    Bs[(kk + 1) * DIM + tid] = v.y;
                    Bs[(kk + 2) * DIM + tid] = v.z;
                    Bs[(kk + 3) * DIM + tid] = v.w;
                }
            }
            __syncthreads();

            // ---- 8 k-steps of WMMA from LDS ----
            for (int ks = 0; ks < 32; ks += 4) {
                v2f a;
                a.x = panel[wave][mA][ks0 + ks + kHalf];
                a.y = panel[wave][mA][ks0 + ks + kHalf + 1];
                const int kk0 = ks + kHalf;
#pragma unroll
                for (int nt = 0; nt < 8; ++nt) {
                    const int ncol = nt * 16 + (lane & 15);
                    v2f b;
                    b.x = Bs[kk0 * DIM + ncol];
                    b.y = Bs[(kk0 + 1) * DIM + ncol];
                    c[nt] = __builtin_amdgcn_wmma_f32_16x16x4_f32(
                        false, a, false, b, (short)0, c[nt], false, false);
                }
            }
        }
    }

    // ---- epilogue: bias (+ReLU) and store ----
    const int mOff = (lane >> 4) << 3;   // 0 or 8
#pragma unroll
    for (int nt = 0; nt < 8; ++nt) {
        const int colg = nt * 16 + (lane & 15);
        const float bv = bias[colg];
#pragma unroll
        for (int i = 0; i < 8; ++i) {
            int node = row0 + i + mOff;
            float val = c[nt][i] + bv;
            if (relu) val = fmaxf(val, 0.f);
            if (node < NNODES) hout[(size_t)node * DIM + colg] = val;
        }
    }
}

// ---------------------------------------------------------------------------
// Kernel 3: single-query MHA for one batch row per block (128 threads).
// dh = 16, scale = 0.25.  Everything staged in LDS.
// ---------------------------------------------------------------------------
__global__ __launch_bounds__(128)
void attn_kernel(const float* __restrict__ h,      // [NNODES, DIM]
                 const int*   __restrict__ batch,  // [BATCHN, BROW]
                 int col0, int L,
                 const float* __restrict__ Wi, const float* __restrict__ bi,
                 const float* __restrict__ Wo, const float* __restrict__ bo,
                 float*       __restrict__ outbuf) // [BATCHN, DIM]
{
    const int b   = blockIdx.x;
    const int tid = threadIdx.x;

    __shared__ float q[DIM];
    __shared__ float Qv[DIM];
    __shared__ float kv[MAXL][DIM];
    __shared__ float Km[MAXL][DIM];
    __shared__ float Vm[MAXL][DIM];
    __shared__ float sc[NHEADS][MAXL];
    __shared__ float aw[NHEADS][MAXL];
    __shared__ float ov[DIM];
    __shared__ int   msk[MAXL];

    const int* brow = batch + (size_t)b * BROW;
    const int vid = brow[0];
    q[tid] = h[(size_t)vid * DIM + tid];
    for (int l = 0; l < L; ++l) {
        int idx = brow[col0 + l];
        if (tid == 0) msk[l] = (idx == 0);
        kv[l][tid] = h[(size_t)idx * DIM + tid];
    }
    __syncthreads();

    // Q projection (row tid of Wq = Wi[0:128])
    {
        float s = bi[tid];
        const float* w = Wi + (size_t)tid * DIM;
        for (int k = 0; k < DIM; ++k) s += q[k] * w[k];
        Qv[tid] = s;
    }
    // K and V projections
    for (int l = 0; l < L; ++l) {
        float skk = bi[DIM + tid];
        float svv = bi[2 * DIM + tid];
        const float* wk = Wi + (size_t)(DIM + tid) * DIM;
        const float* wv = Wi + (size_t)(2 * DIM + tid) * DIM;
        for (int k = 0; k < DIM; ++k) {
            float xv = kv[l][k];
            skk += xv * wk[k];
            svv += xv * wv[k];
        }
        Km[l][tid] = skk;
        Vm[l][tid] = svv;
    }
    __syncthreads();

    // scores
    for (int p = tid; p < NHEADS * L; p += 128) {
        int hh = p / L, l = p % L;
        float s = 0.f;
#pragma unroll
        for (int t = 0; t < 16; ++t) s += Qv[hh * 16 + t] * Km[l][hh * 16 + t];
        s *= 0.25f;                        // 1/sqrt(dh), dh=16
        if (msk[l]) s = -1.0e9f;
        sc[hh][l] = s;
    }
    __syncthreads();

    // softmax per head
    if (tid < NHEADS) {
        float mx = -3.0e38f;
        for (int l = 0; l < L; ++l) mx = fmaxf(mx, sc[tid][l]);
        float sum = 0.f;
        for (int l = 0; l < L; ++l) { float e = __expf(sc[tid][l] - mx); aw[tid][l] = e; sum += e; }
        float inv = 1.0f / sum;
        for (int l = 0; l < L; ++l) aw[tid][l] *= inv;
    }
    __syncthreads();

    // attend
    {
        int hh = tid >> 4;
        float s = 0.f;
        for (int l = 0; l < L; ++l) s += aw[hh][l] * Vm[l][tid];
        ov[tid] = s;
    }
    __syncthreads();

    // output projection
    {
        float s = bo[tid];
        const float* w = Wo + (size_t)tid * DIM;
        for (int k = 0; k < DIM; ++k) s += ov[k] * w[k];
        outbuf[(size_t)b * DIM + tid] = s;
    }
}

// ---------------------------------------------------------------------------
// Kernel 4: FFNN (384 -> 192 -> 1) for pos and neg + BCE-with-logits terms.
// One block (192 threads) per batch row.
// ---------------------------------------------------------------------------
__global__ __launch_bounds__(192)
void ffnn_loss(const float* __restrict__ left, const float* __restrict__ right,
               const float* __restrict__ h, const int* __restrict__ batch,
               const float* __restrict__ fc1W, const float* __restrict__ fc1b,
               const float* __restrict__ fc2W, const float* __restrict__ fc2b,
               float* __restrict__ loss_acc)
{
    const int b   = blockIdx.x;
    const int tid = threadIdx.x;   // 0..191
    __shared__ float xb[3 * DIM];
    __shared__ float hr[192];
    const int* brow = batch + (size_t)b * BROW;

    for (int which = 0; which < 2; ++which) {
        int node = brow[1 + which];          // pos_idx / neg_idx
        if (tid < DIM) {
            xb[tid]           = left[(size_t)b * DIM + tid];
            xb[DIM + tid]     = right[(size_t)b * DIM + tid];
            xb[2 * DIM + tid] = h[(size_t)node * DIM + tid];
        }
        __syncthreads();

        float s = fc1b[tid];
        const float* w = fc1W + (size_t)tid * (3 * DIM);
        for (int k = 0; k < 3 * DIM; ++k) s += xb[k] * w[k];
        s = fmaxf(s, 0.f);
        hr[tid] = s * fc2W[tid];
        __syncthreads();

        if (tid == 0) {
            float z = fc2b[0];
            for (int k = 0; k < 192; ++k) z += hr[k];
            float t = (which == 0) ? 1.f : 0.f;
            float l = fmaxf(z, 0.f) - z * t + log1pf(__expf(-fabsf(z)));
            atomicAdd(loss_acc, l);
        }
        __syncthreads();
    }
}

__global__ void finalize_loss(const float* __restrict__ acc, float* __restrict__ out)
{
    if (threadIdx.x == 0 && blockIdx.x == 0) out[0] = acc[0] / (2.0f * BATCHN);
}

// ---------------------------------------------------------------------------
extern "C" void kernel_launch(void* const* d_in, const int* in_sizes, int n_in,
                              void* d_out, int out_size, void* d_ws, size_t ws_size,
                              hipStream_t stream)
{
    const int*   batch    = (const int*)  d_in[0];
    const int*   eidx     = (const int*)  d_in[1];
    const int*   etype    = (const int*)  d_in[2];
    const float* node_emb = (const float*)d_in[3];
    const float* Wrel1    = (const float*)d_in[4];
    const float* Wroot1   = (const float*)d_in[5];
    const float* b1       = (const float*)d_in[6];
    const float* Wrel2    = (const float*)d_in[7];
    const float* Wroot2   = (const float*)d_in[8];
    const float* b2       = (const float*)d_in[9];
    const float* lWi = (const float*)d_in[10]; const float* lbi = (const float*)d_in[11];
    const float* lWo = (const float*)d_in[12]; const float* lbo = (const float*)d_in[13];
    const float* rWi = (const float*)d_in[14]; const float* rbi = (const float*)d_in[15];
    const float* rWo = (const float*)d_in[16]; const float* rbo = (const float*)d_in[17];
    const float* fc1W = (const float*)d_in[18]; const float* fc1b = (const float*)d_in[19];
    const float* fc2W = (const float*)d_in[20]; const float* fc2b = (const float*)d_in[21];
    float* out = (float*)d_out;
    const int E = in_sizes[2];                 // edge_type element count

    // -------- workspace layout --------
    char* ws = (char*)d_ws;
    size_t off = 0;
    auto wsalloc = [&](size_t bytes) -> void* {
        void* p = ws + off;
        off = (off + bytes + 255) & ~(size_t)255;
        return p;
    };
    const size_t accBytes = sizeof(float) * (size_t)NREL * NNODES * DIM;  // 204.8 MB
    const size_t cntBytes = sizeof(float) * (size_t)NREL * NNODES;
    float* acc   = (float*)wsalloc(accBytes);
    float* cnt   = (float*)wsalloc(cntBytes);
    float* h1    = (float*)wsalloc(sizeof(float) * (size_t)NNODES * DIM);
    float* h2    = (float*)wsalloc(sizeof(float) * (size_t)NNODES * DIM);
    float* left  = (float*)wsalloc(sizeof(float) * (size_t)BATCHN * DIM);
    float* right = (float*)wsalloc(sizeof(float) * (size_t)BATCHN * DIM);
    float* lacc  = (float*)wsalloc(sizeof(float));

    const int scatterBlocks = (E + 7) / 8;            // 8 waves (edges) / 256-thr block
    const int gemmBlocks    = (NNODES + 63) / 64;     // 4 waves x 16 nodes / block

    // -------- layer 1 --------
    (void)hipMemsetAsync(acc, 0, accBytes, stream);
    (void)hipMemsetAsync(cnt, 0, cntBytes, stream);
    rgcn_scatter<<<scatterBlocks, 256, 0, stream>>>(node_emb, eidx, etype, acc, cnt, E);
    rgcn_gemm<<<gemmBlocks, 128, 0, stream>>>(node_emb, acc, cnt, Wrel1, Wroot1, b1, h1, 1);

    // -------- layer 2 --------
    (void)hipMemsetAsync(acc, 0, accBytes, stream);
    (void)hipMemsetAsync(cnt, 0, cntBytes, stream);
    rgcn_scatter<<<scatterBlocks, 256, 0, stream>>>(h1, eidx, etype, acc, cnt, E);
    rgcn_gemm<<<gemmBlocks, 128, 0, stream>>>(h1, acc, cnt, Wrel2, Wroot2, b2, h2, 0);

    // -------- attention: left = cosponser (cols 33..57, L=25), right = subject (cols 3..32, L=30)
    attn_kernel<<<BATCHN, 128, 0, stream>>>(h2, batch, 3 + NSUBJ, NCOSP, lWi, lbi, lWo, lbo, left);
    attn_kernel<<<BATCHN, 128, 0, stream>>>(h2, batch, 3,         NSUBJ, rWi, rbi, rWo, rbo, right);

    // -------- FFNN + loss --------
    (void)hipMemsetAsync(lacc, 0, sizeof(float), stream);
    ffnn_loss<<<BATCHN, 192, 0, stream>>>(left, right, h2, batch, fc1W, fc1b, fc2W, fc2b, lacc);
    finalize_loss<<<1, 32, 0, stream>>>(lacc, out);
}